// screen_shoot_7705171329344
// MI455X (gfx1250) — compile-verified
//
#include <hip/hip_runtime.h>
#include <math.h>

#define BATCH 32
#define HH 512
#define WW 512
#define HWSZ (HH*WW)
#define NPIX (BATCH*3*HWSZ)

typedef __attribute__((ext_vector_type(2))) float v2f;
typedef __attribute__((ext_vector_type(8))) float v8f;

// ---------------------------------------------------------------------------
// Kernel 0: per-batch homography. Solve 8x8 DLT (src->dst) with partial
// pivoting in f64, then 3x3 adjugate inverse. One thread per batch.
// minv layout: 12 floats per batch (9 used, row-major Minv[r][c] at r*3+c).
// ---------------------------------------------------------------------------
__global__ void homog_kernel(const float* __restrict__ dst_offsets,
                             float* __restrict__ minv) {
    int b = threadIdx.x;
    if (b >= BATCH) return;
    const double S = 512.0;
    double xs[4] = {0.0, 0.0, S, S};
    double ys[4] = {0.0, S, 0.0, S};
    double u[4], v[4];
    for (int p = 0; p < 4; ++p) {
        u[p] = xs[p] + (double)dst_offsets[(b*4 + p)*2 + 0];
        v[p] = ys[p] + (double)dst_offsets[(b*4 + p)*2 + 1];
    }
    double A[8][9];
    for (int p = 0; p < 4; ++p) {
        double r0[9] = { xs[p], ys[p], 1.0, 0.0, 0.0, 0.0, -u[p]*xs[p], -u[p]*ys[p], u[p] };
        double r1[9] = { 0.0, 0.0, 0.0, xs[p], ys[p], 1.0, -v[p]*xs[p], -v[p]*ys[p], v[p] };
        for (int c = 0; c < 9; ++c) { A[p][c] = r0[c]; A[p+4][c] = r1[c]; }
    }
    // Gaussian elimination with partial pivoting
    for (int k = 0; k < 8; ++k) {
        int piv = k;
        double best = fabs(A[k][k]);
        for (int r = k+1; r < 8; ++r) {
            double a = fabs(A[r][k]);
            if (a > best) { best = a; piv = r; }
        }
        if (piv != k) {
            for (int c = k; c < 9; ++c) { double t = A[k][c]; A[k][c] = A[piv][c]; A[piv][c] = t; }
        }
        double inv_p = 1.0 / A[k][k];
        for (int r = k+1; r < 8; ++r) {
            double f = A[r][k] * inv_p;
            for (int c = k; c < 9; ++c) A[r][c] -= f * A[k][c];
        }
    }
    double h[9];
    for (int k = 7; k >= 0; --k) {
        double s = A[k][8];
        for (int c = k+1; c < 8; ++c) s -= A[k][c] * h[c];
        h[k] = s / A[k][k];
    }
    h[8] = 1.0;
    // 3x3 inverse via adjugate
    double m0=h[0],m1=h[1],m2=h[2],m3=h[3],m4=h[4],m5=h[5],m6=h[6],m7=h[7],m8=h[8];
    double det = m0*(m4*m8-m5*m7) - m1*(m3*m8-m5*m6) + m2*(m3*m7-m4*m6);
    double id = 1.0/det;
    double inv[9] = {
        (m4*m8-m5*m7)*id, (m2*m7-m1*m8)*id, (m1*m5-m2*m4)*id,
        (m5*m6-m3*m8)*id, (m0*m8-m2*m6)*id, (m2*m3-m0*m5)*id,
        (m3*m7-m4*m6)*id, (m1*m6-m0*m7)*id, (m0*m4-m1*m3)*id };
    for (int k = 0; k < 9; ++k) minv[b*12 + k] = (float)inv[k];
}

// ---------------------------------------------------------------------------
// Kernel 1: WMMA rowbase precompute. For each (b,i):
//   base_r(i) = Minv[r][1]*i + Minv[r][2]   (r = 0..2)
// via two chained V_WMMA_F32_16X16X4_F32:
//   D1 = A1 x B1      A1 row m = [Minv[m][1],0,0,0] (m<3), B1[k][n] = i_n (all k)
//   D  = A2 x B2 + D1 A2 row m = [Minv[m][2],0,0,0] (m<3), B2[k][n] = 1   (all k)
// B is K-uniform and A uses one slot per row, so the result depends only on
// A's row-sum -> immune to any K-slot permutation in the B layout.
// One wave per batch; rowbase stored as float4 (base0,base1,base2,0) per (b,i).
// ---------------------------------------------------------------------------
__global__ __launch_bounds__(32) void rowbase_kernel(const float* __restrict__ minv,
                                                     float* __restrict__ rowbase) {
    int b = blockIdx.x;
    int lane = threadIdx.x;
    int n = lane & 15;
    const float* Mi = minv + b*12;
    float a1v = (lane < 3) ? Mi[lane*3 + 1] : 0.0f;   // lanes 16-31 -> 0 (K=2 col zero)
    float a2v = (lane < 3) ? Mi[lane*3 + 2] : 0.0f;
    v2f a1 = {a1v, 0.0f};
    v2f a2 = {a2v, 0.0f};
    v2f bones = {1.0f, 1.0f};
    for (int i0 = 0; i0 < HH; i0 += 16) {
        float fi = (float)(i0 + n);
        v2f b1 = {fi, fi};                             // all 4 K rows = i_n
        v8f acc = {};
        acc = __builtin_amdgcn_wmma_f32_16x16x4_f32(
            /*neg_a=*/false, a1, /*neg_b=*/false, b1,
            /*c_mod=*/(short)0, acc, /*reuse_a=*/false, /*reuse_b=*/false);
        acc = __builtin_amdgcn_wmma_f32_16x16x4_f32(
            false, a2, false, bones, (short)0, acc, false, false);
        if (lane < 16) {
            float4 o = make_float4(acc[0], acc[1], acc[2], 0.0f);
            ((float4*)rowbase)[b*HH + i0 + lane] = o;
        }
    }
}

// ---------------------------------------------------------------------------
// Kernel 2: batch-independent additive field (light spot + moire), (3,H,W).
// addf[c][i][j] = light(i,j) + 0.2*(min(z1,z2)*2-1)
// ---------------------------------------------------------------------------
__global__ __launch_bounds__(256) void field_kernel(const int* __restrict__ light_xy,
                                                    const float* __restrict__ theta,
                                                    const float* __restrict__ center,
                                                    float* __restrict__ addf) {
    int idx = blockIdx.x*256 + threadIdx.x;     // 0 .. HWSZ-1
    int i = idx >> 9;
    int j = idx & (WW-1);
    const float S = 512.0f;
    const float PI_ = 3.14159f;
    float xf = (float)light_xy[0];
    float yf = (float)light_xy[1];
    float d0 = sqrtf(xf*xf + yf*yf);
    float d1 = sqrtf((xf-S)*(xf-S) + yf*yf);
    float d2 = sqrtf(xf*xf + (yf-S)*(yf-S));
    float d3 = sqrtf((xf-S)*(xf-S) + (yf-S)*(yf-S));
    float ml = fmaxf(fmaxf(d0, d1), fmaxf(d2, d3)) * 0.5f;
    float fi = (float)i, fj = (float)j;
    float l = sqrtf((fi-xf)*(fi-xf) + (fj-yf)*(fj-yf));
    float light = (l < ml) ? (1.0f - l/ml) : 0.0f;
    for (int c = 0; c < 3; ++c) {
        float dx = fi - center[c*2 + 0];
        float dy = fj - center[c*2 + 1];
        float z1 = 0.5f + 0.5f*cosf(2.0f*PI_*sqrtf(dx*dx + dy*dy));
        float th = (theta[c] / 180.0f) * PI_;
        float z2 = 0.5f + 0.5f*cosf(cosf(th)*fj + sinf(th)*fi);
        float m = fminf(z1, z2)*2.0f - 1.0f;
        addf[c*HWSZ + idx] = light + 0.2f*m;
    }
}

// ---------------------------------------------------------------------------
// Kernel 3: fused warp + color + field + noise. 1 thread per (b,i,j), all 3
// channels in-thread (warp coords computed once). Bandwidth-bound.
// ---------------------------------------------------------------------------
__global__ __launch_bounds__(256) void main_kernel(const float* __restrict__ imgs,
                                                   const float* __restrict__ noise,
                                                   const float* __restrict__ rowbase,
                                                   const float* __restrict__ minv,
                                                   const float* __restrict__ addf,
                                                   const float* __restrict__ hue,
                                                   const float* __restrict__ bright,
                                                   const float* __restrict__ contrast,
                                                   float* __restrict__ out) {
    int j = blockIdx.x*256 + threadIdx.x;
    int i = blockIdx.y;
    int b = blockIdx.z;

    const float4 rb = ((const float4*)rowbase)[b*HH + i];
    const float* Mi = minv + b*12;
    float spx = rb.x + (float)j * Mi[0];
    float spy = rb.y + (float)j * Mi[3];
    float spw = rb.z + (float)j * Mi[6];
    float sx = spx / spw;
    float sy = spy / spw;

    float x0f = floorf(sx), y0f = floorf(sy);
    float wx = sx - x0f, wy = sy - y0f;
    int x0 = (int)x0f, y0 = (int)y0f;
    int x1 = x0 + 1,   y1 = y0 + 1;
    float vx0 = (x0 >= 0 && x0 < WW) ? 1.0f : 0.0f;
    float vx1 = (x1 >= 0 && x1 < WW) ? 1.0f : 0.0f;
    float vy0 = (y0 >= 0 && y0 < HH) ? 1.0f : 0.0f;
    float vy1 = (y1 >= 0 && y1 < HH) ? 1.0f : 0.0f;
    int cx0 = min(max(x0, 0), WW-1);
    int cx1 = min(max(x1, 0), WW-1);
    int cy0 = min(max(y0, 0), HH-1);
    int cy1 = min(max(y1, 0), HH-1);
    float w00 = (1.0f-wy)*(1.0f-wx) * (vy0*vx0);
    float w01 = (1.0f-wy)*wx        * (vy0*vx1);
    float w10 = wy*(1.0f-wx)        * (vy1*vx0);
    float w11 = wy*wx               * (vy1*vx1);
    int o00 = cy0*WW + cx0, o01 = cy0*WW + cx1;
    int o10 = cy1*WW + cx0, o11 = cy1*WW + cx1;

    float K = contrast[b];
    float Br = bright[b];
    const float* ib = imgs + (size_t)(b*3)*HWSZ;

    float r[3];
    #pragma unroll
    for (int c = 0; c < 3; ++c) {
        const float* p = ib + (size_t)c*HWSZ;
        float v = w00*p[o00] + w01*p[o01] + w10*p[o10] + w11*p[o11];
        v = v*K + hue[b*3 + c];
        v = v + Br;
        v = fminf(1.0f, fmaxf(-1.0f, v));
        r[c] = v;
    }
    float lum = (0.3f*r[0] + 0.6f*r[1] + 0.1f*r[2]) * (1.0f/3.0f);

    int pix = i*WW + j;
    #pragma unroll
    for (int c = 0; c < 3; ++c) {
        float v = 0.7f*r[c] + 0.3f*lum;
        v += addf[c*HWSZ + pix];
        v += 0.03162277660168379f * noise[(size_t)(b*3 + c)*HWSZ + pix];
        out[(size_t)(b*3 + c)*HWSZ + pix] = v;
    }
}

// ---------------------------------------------------------------------------
extern "C" void kernel_launch(void* const* d_in, const int* in_sizes, int n_in,
                              void* d_out, int out_size, void* d_ws, size_t ws_size,
                              hipStream_t stream) {
    const float* imgs        = (const float*)d_in[0];
    const float* cover       = (const float*)d_in[1];
    const float* dst_offsets = (const float*)d_in[2];
    const float* hue_shift   = (const float*)d_in[3];
    const float* bright      = (const float*)d_in[4];
    const float* contrast    = (const float*)d_in[5];
    const int*   light_xy    = (const int*)d_in[6];
    const float* moire_theta = (const float*)d_in[7];
    const float* moire_center= (const float*)d_in[8];
    const float* noise       = (const float*)d_in[9];
    float* out = (float*)d_out;

    // workspace partitions (all 16B aligned)
    float* minv    = (float*)d_ws;                           // 32*12 floats = 1536 B
    float* rowbase = (float*)((char*)d_ws + 1536);           // 32*512*4 floats = 256 KB
    float* addf    = (float*)((char*)d_ws + 1536 + 262144);  // 3*HW floats = 3 MB

    homog_kernel<<<1, 32, 0, stream>>>(dst_offsets, minv);
    rowbase_kernel<<<BATCH, 32, 0, stream>>>(minv, rowbase);
    field_kernel<<<HWSZ/256, 256, 0, stream>>>(light_xy, moire_theta, moire_center, addf);

    dim3 grid(WW/256, HH, BATCH);
    main_kernel<<<grid, 256, 0, stream>>>(imgs, noise, rowbase, minv, addf,
                                          hue_shift, bright, contrast, out);

    // second output: cover passthrough
    hipMemcpyAsync(out + (size_t)NPIX, cover, (size_t)NPIX * sizeof(float),
                   hipMemcpyDeviceToDevice, stream);
}